// PointGraphTransformerA_9612136808566
// MI455X (gfx1250) — compile-verified
//
#include <hip/hip_runtime.h>

typedef float v2f __attribute__((ext_vector_type(2)));
typedef float v8f __attribute__((ext_vector_type(8)));

#define FPS_N     65536
#define FPS_M     2048          // FPS_N / 32
#define KNN_K     128
#define SEL_CPB   16            // centroids per block (one WMMA A tile)
#define SEL_THREADS 256
#define SEL_WAVES 8
#define NBUCKET   2048          // 11-bit monotonic float key
#define CAND_CAP  512

// ---------------------------------------------------------------------------
// Kernel 1: farthest point sampling. One workgroup (32 wave32s on one WGP),
// min-dist lives in registers (64 floats/lane), argmax via packed u64 max.
// ---------------------------------------------------------------------------
__global__ __launch_bounds__(1024)
void fps_kernel(const float* __restrict__ pos, int* __restrict__ centIdx) {
    __shared__ unsigned long long s_red[32];
    __shared__ unsigned s_cur;
    const int tid  = threadIdx.x;
    const int lane = tid & 31;
    const int wid  = tid >> 5;

    float dist[64];
#pragma unroll
    for (int j = 0; j < 64; ++j) dist[j] = __builtin_inff();

    if (tid == 0) { s_cur = 0u; centIdx[0] = 0; }
    __syncthreads();

    for (int t = 1; t < FPS_M; ++t) {
        const unsigned cur = s_cur;
        const float cx = pos[3*cur+0], cy = pos[3*cur+1], cz = pos[3*cur+2];
        unsigned long long best = 0ull;
#pragma unroll
        for (int j = 0; j < 64; ++j) {
            const int i = tid + (j << 10);
            const float dx = pos[3*i+0] - cx;
            const float dy = pos[3*i+1] - cy;
            const float dz = pos[3*i+2] - cz;
            const float d  = fmaf(dx, dx, fmaf(dy, dy, dz * dz));
            const float nd = fminf(dist[j], d);
            dist[j] = nd;
            // bigger dist wins; tie -> smaller index (argmax picks first max)
            const unsigned long long pk =
                ((unsigned long long)__float_as_uint(nd) << 32) |
                (unsigned long long)(0xFFFFFFFFu - (unsigned)i);
            best = pk > best ? pk : best;
        }
        // wave32 reduction
#pragma unroll
        for (int off = 16; off > 0; off >>= 1) {
            const unsigned long long o = __shfl_xor(best, off, 32);
            best = o > best ? o : best;
        }
        if (lane == 0) s_red[wid] = best;
        __syncthreads();
        if (wid == 0) {
            unsigned long long b = s_red[lane];
#pragma unroll
            for (int off = 16; off > 0; off >>= 1) {
                const unsigned long long o = __shfl_xor(b, off, 32);
                b = o > b ? o : b;
            }
            if (lane == 0) {
                const unsigned win = 0xFFFFFFFFu - (unsigned)(b & 0xFFFFFFFFull);
                s_cur = win;
                centIdx[t] = (int)win;
            }
        }
        __syncthreads();
    }
}

// ---------------------------------------------------------------------------
// Kernel 2: per-point |p|^2 and gathered centroid coords + |c|^2.
// ---------------------------------------------------------------------------
__global__ void prep_kernel(const float* __restrict__ pos, const int* __restrict__ centIdx,
                            float* __restrict__ cx, float* __restrict__ cy,
                            float* __restrict__ cz, float* __restrict__ cc,
                            float* __restrict__ pp) {
    const int i = blockIdx.x * blockDim.x + threadIdx.x;
    if (i < FPS_N) {
        const float x = pos[3*i], y = pos[3*i+1], z = pos[3*i+2];
        pp[i] = fmaf(x, x, fmaf(y, y, z * z));
    }
    if (i < FPS_M) {
        const int id = centIdx[i];
        const float x = pos[3*id], y = pos[3*id+1], z = pos[3*id+2];
        cx[i] = x; cy[i] = y; cz[i] = z;
        cc[i] = fmaf(x, x, fmaf(y, y, z * z));
    }
}

// ---------------------------------------------------------------------------
// Kernel 3: per-block 16 centroids; WMMA 16x16x4 f32 computes C.P^T tiles;
// histogram-select top-128 per centroid, bitonic sort, gather output.
// Dynamic LDS: 16*2048 u32 histogram, reused as 16*512 u64 candidate lists.
// ---------------------------------------------------------------------------
__global__ __launch_bounds__(SEL_THREADS)
void knn_select_kernel(const float* __restrict__ pos,
                       const float* __restrict__ cxv, const float* __restrict__ cyv,
                       const float* __restrict__ czv, const float* __restrict__ ccv,
                       const float* __restrict__ pp,
                       float* __restrict__ out) {
    extern __shared__ unsigned char smem[];
    unsigned*           hist = (unsigned*)smem;            // 16*2048 u32 = 128KB
    unsigned long long* cand = (unsigned long long*)smem;  // 16*512  u64 =  64KB (reuse)
    __shared__ float    s_cx[16], s_cy[16], s_cz[16], s_cc[16];
    __shared__ unsigned s_T[16], s_cnt[16];

    const int tid  = threadIdx.x;
    const int lane = tid & 31;
    const int w    = tid >> 5;
    const int g0   = blockIdx.x * SEL_CPB;

    if (tid < 16) {
        s_cx[tid] = cxv[g0+tid]; s_cy[tid] = cyv[g0+tid];
        s_cz[tid] = czv[g0+tid]; s_cc[tid] = ccv[g0+tid];
    }
    for (int i = tid; i < SEL_CPB * NBUCKET; i += SEL_THREADS) hist[i] = 0u;
    __syncthreads();

    // A tile: 16x4 f32 layout -> lanes 0-15 hold (K=0,K=1)=(x,y); lanes 16-31 (K=2,K=3)=(z,0)
    const int  l15 = lane & 15;
    const bool lo  = (lane < 16);
    const int  sel = (lane >> 4) << 1;      // 0 for lanes 0-15, 2 for lanes 16-31
    v2f a;
    a[0] = lo ? s_cx[l15] : s_cz[l15];
    a[1] = lo ? s_cy[l15] : 0.f;
    const int mBase = lo ? 0 : 8;           // C/D rows owned by this lane half
    float ccr[8];
#pragma unroll
    for (int v = 0; v < 8; ++v) ccr[v] = s_cc[mBase + v];

    const int ntile = FPS_N / 16;

    // ---- Pass 1: histogram of 11-bit monotonic keys ----
    for (int tile = w; tile < ntile; tile += SEL_WAVES) {
        const int n = tile * 16 + l15;
        // branchless B tile: lane loads (x or z) and y; upper half's second elem -> 0
        const float e0 = pos[3*n + sel];
        const float e1 = pos[3*n + 1];
        v2f b; b[0] = e0; b[1] = lo ? e1 : 0.f;
        v8f acc = {};
        acc = __builtin_amdgcn_wmma_f32_16x16x4_f32(false, a, false, b,
                                                    (short)0, acc, false, false);
        const float ppn = pp[n];
#pragma unroll
        for (int v = 0; v < 8; ++v) {
            float dd = fmaf(-2.f, acc[v], ccr[v] + ppn);
            dd = fmaxf(dd, 0.f);
            const unsigned key = __float_as_uint(dd) >> 20;   // < 2048 for finite dd>=0
            atomicAdd(&hist[(mBase + v) * NBUCKET + key], 1u);
        }
    }
    __syncthreads();

    // ---- Threshold: smallest key with cumulative count >= K ----
    if (tid < 16) {
        unsigned run = 0, T = NBUCKET - 1;
        for (int k = 0; k < NBUCKET; ++k) {
            run += hist[tid * NBUCKET + k];
            if (run >= KNN_K) { T = (unsigned)k; break; }
        }
        s_T[tid] = T;
        s_cnt[tid] = 0u;
    }
    __syncthreads();

    // hoist per-row thresholds into registers
    unsigned tr[8];
#pragma unroll
    for (int v = 0; v < 8; ++v) tr[v] = s_T[mBase + v];

    // ---- Pass 2: collect candidates (hist LDS reused as cand lists) ----
    for (int tile = w; tile < ntile; tile += SEL_WAVES) {
        const int n = tile * 16 + l15;
        const float e0 = pos[3*n + sel];
        const float e1 = pos[3*n + 1];
        v2f b; b[0] = e0; b[1] = lo ? e1 : 0.f;
        v8f acc = {};
        acc = __builtin_amdgcn_wmma_f32_16x16x4_f32(false, a, false, b,
                                                    (short)0, acc, false, false);
        const float ppn = pp[n];
#pragma unroll
        for (int v = 0; v < 8; ++v) {
            float dd = fmaf(-2.f, acc[v], ccr[v] + ppn);
            dd = fmaxf(dd, 0.f);
            const int m = mBase + v;
            const unsigned key = __float_as_uint(dd) >> 20;
            if (key <= tr[v]) {
                const unsigned slot = atomicAdd(&s_cnt[m], 1u);
                if (slot < CAND_CAP)
                    cand[m * CAND_CAP + slot] =
                        ((unsigned long long)__float_as_uint(dd) << 32) |
                        (unsigned long long)(unsigned)n;
            }
        }
    }
    __syncthreads();

    // ---- Pad unused slots with +inf keys ----
    for (int s = tid; s < SEL_CPB * CAND_CAP; s += SEL_THREADS) {
        const int m = s >> 9;
        const unsigned k = (unsigned)(s & (CAND_CAP - 1));
        if (k >= s_cnt[m]) cand[s] = ~0ull;
    }
    __syncthreads();

    // ---- Bitonic sort: wave w sorts lists 2w, 2w+1 (same-wave LDS is in-order) ----
    for (int c = 0; c < 2; ++c) {
        unsigned long long* A = cand + (2 * w + c) * CAND_CAP;
        for (int kk = 2; kk <= CAND_CAP; kk <<= 1) {
            for (int j = kk >> 1; j > 0; j >>= 1) {
#pragma unroll
                for (int s = 0; s < CAND_CAP / 32; ++s) {
                    const int i2 = lane + (s << 5);
                    const int p  = i2 ^ j;
                    if (p > i2) {
                        const bool up = ((i2 & kk) == 0);
                        const unsigned long long x = A[i2];
                        const unsigned long long y = A[p];
                        if (up ? (x > y) : (x < y)) { A[i2] = y; A[p] = x; }
                    }
                }
            }
        }
    }
    __syncthreads();

    // ---- Output: slot0 = centroid, slots 1..127 = sorted neighbors 1..127 ----
    const int m = tid >> 4;
    const int g = g0 + m;
#pragma unroll
    for (int s = 0; s < 8; ++s) {
        const int k = (tid & 15) + (s << 4);
        float x, y, z;
        if (k == 0) { x = s_cx[m]; y = s_cy[m]; z = s_cz[m]; }
        else {
            const unsigned idx = (unsigned)(cand[m * CAND_CAP + k] & 0xFFFFFFFFull);
            x = pos[3*idx]; y = pos[3*idx+1]; z = pos[3*idx+2];
        }
        const int o = (g * KNN_K + k) * 3;
        out[o+0] = x; out[o+1] = y; out[o+2] = z;
    }
}

// ---------------------------------------------------------------------------
extern "C" void kernel_launch(void* const* d_in, const int* in_sizes, int n_in,
                              void* d_out, int out_size, void* d_ws, size_t ws_size,
                              hipStream_t stream) {
    (void)in_sizes; (void)n_in; (void)out_size; (void)ws_size;
    const float* pos = (const float*)d_in[0];
    float* out = (float*)d_out;
    char* ws = (char*)d_ws;
    int*   centIdx = (int*)ws;                       //  8 KB
    float* cx = (float*)(ws + 4  * FPS_M);
    float* cy = (float*)(ws + 8  * FPS_M);
    float* cz = (float*)(ws + 12 * FPS_M);
    float* cc = (float*)(ws + 16 * FPS_M);
    float* pp = (float*)(ws + 20 * FPS_M);           // 256 KB

    fps_kernel<<<1, 1024, 0, stream>>>(pos, centIdx);
    prep_kernel<<<(FPS_N + 255) / 256, 256, 0, stream>>>(pos, centIdx, cx, cy, cz, cc, pp);
    knn_select_kernel<<<FPS_M / SEL_CPB, SEL_THREADS,
                        SEL_CPB * NBUCKET * sizeof(unsigned), stream>>>(
        pos, cx, cy, cz, cc, pp, out);
}